// SGC_48241072669149
// MI455X (gfx1250) — compile-verified
//
#include <hip/hip_runtime.h>
#include <hip/hip_bf16.h>

// SGC: out = ( relu( P^2(x) @ W1 + b1 ) -> P^2 -> @ W2 + b2 )
// P = D^-1/2 (A + I) D^-1/2, applied K=2 per conv.
// N=100000 nodes, E=3.2M edges, feature dim 128 throughout propagation.

#define NNODE 100000
#define NEDGE 3200000
#define DIM   128

typedef __attribute__((ext_vector_type(2))) float v2f;
typedef __attribute__((ext_vector_type(8))) float v8f;

// ---------------- degree / normalization ----------------

__global__ void deg_init_kernel(float* __restrict__ deg, int n) {
  int i = blockIdx.x * blockDim.x + threadIdx.x;
  if (i < n) deg[i] = 1.0f;              // self-loop weight
}

__global__ void deg_accum_kernel(const int* __restrict__ col,
                                 const float* __restrict__ w,
                                 float* __restrict__ deg, int e) {
  int i = blockIdx.x * blockDim.x + threadIdx.x;
  if (i < e) atomicAdd(&deg[col[i]], w[i]);
}

__global__ void dinv_kernel(float* __restrict__ deg, int n) {
  int i = blockIdx.x * blockDim.x + threadIdx.x;
  if (i < n) {
    float d = deg[i];
    deg[i] = (d > 0.0f) ? rsqrtf(d) : 0.0f;   // becomes dinv in place
  }
}

// Per-edge coefficient, computed ONCE and reused by all 4 SpMM rounds.
__global__ void norm_kernel(const int* __restrict__ row,
                            const int* __restrict__ col,
                            const float* __restrict__ w,
                            const float* __restrict__ dinv,
                            float* __restrict__ norm, int e) {
  int i = blockIdx.x * blockDim.x + threadIdx.x;
  if (i < e) norm[i] = dinv[row[i]] * w[i] * dinv[col[i]];
}

// ---------------- SpMM: out = P @ in (DIM = 128) ----------------

// Self-loop term also initializes `out`:  out[i,:] = dinv[i]^2 * in[i,:]
__global__ void spmm_self_kernel(const float* __restrict__ in,
                                 float* __restrict__ out,
                                 const float* __restrict__ dinv, int n) {
  long long t = (long long)blockIdx.x * blockDim.x + threadIdx.x;  // n*32 threads
  int node = (int)(t >> 5);
  int q    = ((int)t & 31) * 4;
  if (node < n) {
    float s = dinv[node];
    s = s * s;
    const float4 v = *(const float4*)(in + (size_t)node * DIM + q);
    float4 r;
    r.x = v.x * s; r.y = v.y * s; r.z = v.z * s; r.w = v.w * s;
    *(float4*)(out + (size_t)node * DIM + q) = r;
  }
}

// One wave32 per edge: lane moves a contiguous float4 of the 128-f32 row.
// Gather is a b128 load; scatter is 4x global_atomic_add_f32.
__global__ void spmm_edge_kernel(const int* __restrict__ row,
                                 const int* __restrict__ col,
                                 const float* __restrict__ norm,
                                 const float* __restrict__ in,
                                 float* __restrict__ out, int e) {
  long long t = (long long)blockIdx.x * blockDim.x + threadIdx.x;
  int eid  = (int)(t >> 5);
  int lane = (int)t & 31;
  if (eid >= e) return;
  int r = row[eid];
  int c = col[eid];
  float coeff = norm[eid];
  const float4 v = *(const float4*)(in + (size_t)r * DIM + lane * 4);
  float* o = out + (size_t)c * DIM + lane * 4;
  atomicAdd(o + 0, coeff * v.x);
  atomicAdd(o + 1, coeff * v.y);
  atomicAdd(o + 2, coeff * v.z);
  atomicAdd(o + 3, coeff * v.w);
}

// ---------------- dense GEMM with fp32 WMMA ----------------
// out[M,N] = A[M,K] @ W[K,N] + bias[N]  (optional ReLU)
// W (K x N, <=64KB) is staged in LDS by the whole block, then each of the
// blockDim.y waves computes one 16x16 tile via V_WMMA_F32_16X16X4_F32.
// A frag (16x4 f32): lanes 0-15 -> M=lane, K={k,k+1}; lanes 16-31 -> K={k+2,k+3}.
// B frag (4x16 f32): lanes 0-15 -> N=lane, K={k,k+1}; lanes 16-31 -> K={k+2,k+3}.
// C/D (16x16 f32, 8 VGPRs): reg r -> M = r + 8*(lane>=16), N = lane&15.
template <int RELU>
__global__ void gemm_wmma_kernel(const float* __restrict__ A,
                                 const float* __restrict__ W,
                                 const float* __restrict__ bias,
                                 float* __restrict__ out,
                                 int M, int N, int K) {
  extern __shared__ float lds_w[];     // K*N floats

  const int lane = threadIdx.x;        // 0..31, full wave
  const int tn   = threadIdx.y;        // wave id within block = N-tile
  const int tm   = blockIdx.x;         // M-tile
  const int half = lane >> 4;          // 0: K lo pair, 1: K hi pair
  const int l16  = lane & 15;
  const int arow = tm * 16 + l16;
  const int bcol = tn * 16 + l16;

  // ---- stage W into LDS (whole block, float4 copies) ----
  {
    const int tid  = tn * 32 + lane;       // 0 .. 2N-1
    const int nthr = blockDim.y * 32;      // = 2N
    const int nvec = (K * N) >> 2;         // float4 count
    const float4* __restrict__ w4 = (const float4*)W;
    float4* l4 = (float4*)lds_w;
    for (int i = tid; i < nvec; i += nthr) l4[i] = w4[i];
  }
  __syncthreads();

  v8f c = {};
  for (int k = 0; k < K; k += 4) {
    const int ka = k + half * 2;
    // A: two consecutive K values of one row -> 8B aligned pair load
    v2f a = *(const v2f*)(A + (size_t)arow * K + ka);
    v2f b;
    b.x = lds_w[(ka + 0) * N + bcol];
    b.y = lds_w[(ka + 1) * N + bcol];
    c = __builtin_amdgcn_wmma_f32_16x16x4_f32(
        /*neg_a=*/false, a, /*neg_b=*/false, b,
        /*c_mod=*/(short)0, c, /*reuse_a=*/false, /*reuse_b=*/false);
  }

  const float bv = bias[bcol];
#pragma unroll
  for (int r = 0; r < 8; ++r) {
    int m = tm * 16 + r + half * 8;
    float v = c[r] + bv;
    if (RELU) v = fmaxf(v, 0.0f);
    out[(size_t)m * N + bcol] = v;
  }
}

// ---------------- launch ----------------

extern "C" void kernel_launch(void* const* d_in, const int* in_sizes, int n_in,
                              void* d_out, int out_size, void* d_ws, size_t ws_size,
                              hipStream_t stream) {
  (void)in_sizes; (void)n_in; (void)out_size; (void)ws_size;

  const float* x  = (const float*)d_in[0];
  const int*   ei = (const int*)d_in[1];    // (2, E) row-major
  const float* ew = (const float*)d_in[2];
  const float* W1 = (const float*)d_in[3];
  const float* b1 = (const float*)d_in[4];
  const float* W2 = (const float*)d_in[5];
  const float* b2 = (const float*)d_in[6];

  const int n = NNODE;
  const int e = NEDGE;
  const int* row = ei;         // edge_index[0]
  const int* col = ei + e;     // edge_index[1]

  // workspace: dinv[N] | norm[E] | bufA[N*128] | bufB[N*128]  (~116 MB)
  float* dinv = (float*)d_ws;
  float* norm = dinv + (((size_t)n + 63) & ~(size_t)63);
  float* bufA = norm + (((size_t)e + 63) & ~(size_t)63);
  float* bufB = bufA + (size_t)n * DIM;

  const int TB = 256;
  // 1) deg = 1 + segment_sum(w, col);  dinv = rsqrt(deg);  norm per edge
  deg_init_kernel<<<(n + TB - 1) / TB, TB, 0, stream>>>(dinv, n);
  deg_accum_kernel<<<(e + TB - 1) / TB, TB, 0, stream>>>(col, ew, dinv, e);
  dinv_kernel<<<(n + TB - 1) / TB, TB, 0, stream>>>(dinv, n);
  norm_kernel<<<(e + TB - 1) / TB, TB, 0, stream>>>(row, col, ew, dinv, norm, e);

  const long long selfThreads = (long long)n * 32;
  const long long edgeThreads = (long long)e * 32;
  const int selfBlocks = (int)((selfThreads + TB - 1) / TB);
  const int edgeBlocks = (int)((edgeThreads + TB - 1) / TB);

#define SPMM(IN_, OUT_)                                                              \
  do {                                                                               \
    spmm_self_kernel<<<selfBlocks, TB, 0, stream>>>((IN_), (OUT_), dinv, n);         \
    spmm_edge_kernel<<<edgeBlocks, TB, 0, stream>>>(row, col, norm, (IN_),           \
                                                    (OUT_), e);                      \
  } while (0)

  // conv1 propagation (K=2): x -> bufA -> bufB
  SPMM(x, bufA);
  SPMM(bufA, bufB);

  // H = relu(bufB @ W1 + b1) -> bufA   (M=100000 = 6250*16 exact)
  gemm_wmma_kernel<1><<<dim3(n / 16), dim3(32, DIM / 16),
                        DIM * DIM * sizeof(float), stream>>>(
      bufB, W1, b1, bufA, n, DIM, DIM);

  // conv2 propagation (K=2): bufA -> bufB -> bufA
  SPMM(bufA, bufB);
  SPMM(bufB, bufA);

  // out = bufA @ W2 + b2
  gemm_wmma_kernel<0><<<dim3(n / 16), dim3(32, 64 / 16),
                        DIM * 64 * sizeof(float), stream>>>(
      bufA, W2, b2, (float*)d_out, n, 64, DIM);

#undef SPMM
}